// MultiHeadAttention_20813411516788
// MI455X (gfx1250) — compile-verified
//
#include <hip/hip_runtime.h>
#include <hip/hip_bf16.h>
#include <cstdint>

// MI455X (gfx1250) multi-head attention forward, fp32 end-to-end using
// V_WMMA_F32_16X16X4_F32 for all matrix math (projections + QK^T + PV),
// flash-attention style streaming softmax (no SxS score materialization).

typedef __attribute__((ext_vector_type(2))) float v2f;
typedef __attribute__((ext_vector_type(8))) float v8f;

#define D_MODEL 1024
#define N_HEADS 16
#define D_K     64

__device__ __forceinline__ v8f wmma4(v2f a, v2f b, v8f c) {
  // D = A(16x4 f32) * B(4x16 f32) + C(16x16 f32)
  return __builtin_amdgcn_wmma_f32_16x16x4_f32(
      /*neg_a=*/false, a, /*neg_b=*/false, b,
      /*c_mod=*/(short)0, c, /*reuse_a=*/false, /*reuse_b=*/false);
}

// ---------------------------------------------------------------------------
// GEMM: Y[M,N] = X[M,K] * W[N,K]^T + bias[N]   (torch nn.Linear semantics)
// Block tile 128(M) x 128(N), 8 waves in 4x2, each wave 32(M) x 64(N).
// Per k-step: 6 fragment loads feed 8 WMMAs (vs 4/4 before) -> better LDS
// intensity and more independent WMMAs to hide ds_load latency.
// ---------------------------------------------------------------------------
constexpr int GM = 128, GN = 128, GK = 32, GS = GK + 4; // LDS row stride 36

__global__ __launch_bounds__(256) void gemm_xwT_bias_kernel(
    const float* __restrict__ X, const float* __restrict__ W,
    const float* __restrict__ bias, float* __restrict__ Y,
    int M, int N, int K) {
  __shared__ float Xs[GM][GS];
  __shared__ float Ws[GN][GS];

  const int tid  = threadIdx.x;
  const int lane = tid & 31, wave = tid >> 5;
  const int half = lane >> 4, l16 = lane & 15;
  const int wm = wave & 3, wn = wave >> 2;           // 4(M) x 2(N) wave grid
  const int m0 = blockIdx.x * GM, n0 = blockIdx.y * GN;

  v8f acc[2][4] = {};                                 // 32x64 per wave

  for (int k0 = 0; k0 < K; k0 += GK) {
#pragma unroll
    for (int i = 0; i < 4; ++i) {                    // 128x32 X tile
      int idx = i * 256 + tid;
      int r = idx >> 3, c4 = (idx & 7) << 2;
      *(float4*)&Xs[r][c4] =
          *(const float4*)&X[(size_t)(m0 + r) * K + k0 + c4];
    }
#pragma unroll
    for (int i = 0; i < 4; ++i) {                    // 128x32 W tile (row n, col k)
      int idx = i * 256 + tid;
      int r = idx >> 3, c4 = (idx & 7) << 2;
      *(float4*)&Ws[r][c4] =
          *(const float4*)&W[(size_t)(n0 + r) * K + k0 + c4];
    }
    // Prefetch next K-chunk into cache while this chunk computes
    // (global_prefetch_b8: no LOADcnt, pure latency hiding).
    if (k0 + GK < K) {
      int k0n = k0 + GK;
      const float* pf = (tid < 128)
          ? &X[(size_t)(m0 + tid) * K + k0n]
          : &W[(size_t)(n0 + (tid - 128)) * K + k0n];
      __builtin_prefetch(pf, 0, 1);
    }
    __syncthreads();
#pragma unroll
    for (int ks = 0; ks < GK; ks += 4) {
      // A frag: A[m][2h],A[m][2h+1]; B frag: W[n][2h],W[n][2h+1] (= B^T rows)
      v2f a0 = *(const v2f*)&Xs[wm * 32 + l16][ks + 2 * half];
      v2f a1 = *(const v2f*)&Xs[wm * 32 + 16 + l16][ks + 2 * half];
      v2f b0 = *(const v2f*)&Ws[wn * 64 + l16][ks + 2 * half];
      v2f b1 = *(const v2f*)&Ws[wn * 64 + 16 + l16][ks + 2 * half];
      v2f b2 = *(const v2f*)&Ws[wn * 64 + 32 + l16][ks + 2 * half];
      v2f b3 = *(const v2f*)&Ws[wn * 64 + 48 + l16][ks + 2 * half];
      acc[0][0] = wmma4(a0, b0, acc[0][0]);
      acc[0][1] = wmma4(a0, b1, acc[0][1]);
      acc[0][2] = wmma4(a0, b2, acc[0][2]);
      acc[0][3] = wmma4(a0, b3, acc[0][3]);
      acc[1][0] = wmma4(a1, b0, acc[1][0]);
      acc[1][1] = wmma4(a1, b1, acc[1][1]);
      acc[1][2] = wmma4(a1, b2, acc[1][2]);
      acc[1][3] = wmma4(a1, b3, acc[1][3]);
    }
    __syncthreads();
  }

#pragma unroll
  for (int tm = 0; tm < 2; ++tm)
#pragma unroll
    for (int tn = 0; tn < 4; ++tn) {
      int ncol = n0 + wn * 64 + tn * 16 + l16;
      float bb = bias[ncol];
#pragma unroll
      for (int r = 0; r < 8; ++r) {
        int mrow = m0 + wm * 32 + tm * 16 + r + 8 * half; // C layout (ISA 7.12.2)
        Y[(size_t)mrow * N + ncol] = acc[tm][tn][r] + bb;
      }
    }
}

// ---------------------------------------------------------------------------
// Flash attention: one block (4 waves) handles 64 q-rows of one (b,h).
// Wave w owns q-rows [q0+16w, q0+16w+16). Streams 64-wide K/V blocks.
// Mask (67MB) and per-head K/V stay L2-resident (192MB) across re-reads.
// ---------------------------------------------------------------------------
constexpr int AS = 68; // LDS row stride (floats): even (b64), %4==0 (b128), bank-safe

__global__ __launch_bounds__(128) void flash_attn_kernel(
    const float* __restrict__ Qb, const float* __restrict__ Kb,
    const float* __restrict__ Vb, const int* __restrict__ mask,
    float* __restrict__ Ctx, int S) {
  __shared__ float Qs[64][AS];
  __shared__ float Ks[64][AS];
  __shared__ float Vs[64][AS];
  __shared__ float Ps[4][16][AS];   // per-wave P tile (C->A relayout)

  const int tid  = threadIdx.x;
  const int lane = tid & 31, wave = tid >> 5;
  const int half = lane >> 4, l16 = lane & 15;
  const int q0 = blockIdx.x * 64;
  const int b  = blockIdx.y >> 4, h = blockIdx.y & 15;

  const float* Qg = Qb + (size_t)b * S * D_MODEL + h * D_K;
  const float* Kg = Kb + (size_t)b * S * D_MODEL + h * D_K;
  const float* Vg = Vb + (size_t)b * S * D_MODEL + h * D_K;
  const int*   Mg = mask + (size_t)b * S * S;
  float*       Cg = Ctx + (size_t)b * S * D_MODEL + h * D_K;

#pragma unroll
  for (int i = 0; i < 8; ++i) {                      // 64x64 Q tile
    int idx = i * 128 + tid;
    int r = idx >> 4, c4 = (idx & 15) << 2;
    *(float4*)&Qs[r][c4] =
        *(const float4*)&Qg[(size_t)(q0 + r) * D_MODEL + c4];
  }

  v8f o[4] = {};
  float mrun[8], lrun[8];
#pragma unroll
  for (int r = 0; r < 8; ++r) { mrun[r] = -1e38f; lrun[r] = 0.f; }

  for (int kv0 = 0; kv0 < S; kv0 += 64) {
#pragma unroll
    for (int i = 0; i < 8; ++i) {                    // 64x64 K and V tiles
      int idx = i * 128 + tid;
      int r = idx >> 4, c4 = (idx & 15) << 2;
      *(float4*)&Ks[r][c4] =
          *(const float4*)&Kg[(size_t)(kv0 + r) * D_MODEL + c4];
      *(float4*)&Vs[r][c4] =
          *(const float4*)&Vg[(size_t)(kv0 + r) * D_MODEL + c4];
    }
    // Prefetch next K/V block (rows split across the 128 threads).
    if (kv0 + 64 < S) {
      int rn = kv0 + 64 + (tid & 63);
      int co = (tid >> 6) * 32;                      // two 128B halves per row
      __builtin_prefetch(&Kg[(size_t)rn * D_MODEL + co], 0, 1);
      __builtin_prefetch(&Vg[(size_t)rn * D_MODEL + co], 0, 1);
    }
    __syncthreads();

    // S = Q . K^T : 16(q) x 64(k) per wave, reduction over d_k=64
    v8f sc[4] = {};
#pragma unroll
    for (int ks = 0; ks < D_K; ks += 4) {
      v2f a = *(const v2f*)&Qs[wave * 16 + l16][ks + 2 * half];
#pragma unroll
      for (int t = 0; t < 4; ++t) {
        v2f bf = *(const v2f*)&Ks[t * 16 + l16][ks + 2 * half];
        sc[t] = wmma4(a, bf, sc[t]);
      }
    }

    // scale (1/sqrt(64)) + mask (finite NEG_INF, matches reference) + row max
    float rmax[8];
#pragma unroll
    for (int r = 0; r < 8; ++r) rmax[r] = -1e38f;
#pragma unroll
    for (int t = 0; t < 4; ++t) {
      int kcol = kv0 + t * 16 + l16;
#pragma unroll
      for (int r = 0; r < 8; ++r) {
        int qrow = q0 + wave * 16 + r + 8 * half;
        float v = sc[t][r] * 0.125f;
        v = Mg[(size_t)qrow * S + kcol] ? v : -1e30f;
        sc[t][r] = v;
        rmax[r] = fmaxf(rmax[r], v);
      }
    }
#pragma unroll
    for (int r = 0; r < 8; ++r)
#pragma unroll
      for (int off = 1; off < 16; off <<= 1)
        rmax[r] = fmaxf(rmax[r], __shfl_xor(rmax[r], off, 32));

    // online softmax update; write P (exp'd scores) to per-wave LDS tile
    float corr[8], rsum[8];
#pragma unroll
    for (int r = 0; r < 8; ++r) {
      float mnew = fmaxf(mrun[r], rmax[r]);
      corr[r] = __expf(mrun[r] - mnew);
      mrun[r] = mnew;
      rsum[r] = 0.f;
    }
#pragma unroll
    for (int t = 0; t < 4; ++t)
#pragma unroll
      for (int r = 0; r < 8; ++r) {
        float p = __expf(sc[t][r] - mrun[r]);
        rsum[r] += p;
        Ps[wave][r + 8 * half][t * 16 + l16] = p;
      }
#pragma unroll
    for (int r = 0; r < 8; ++r) {
#pragma unroll
      for (int off = 1; off < 16; off <<= 1)
        rsum[r] += __shfl_xor(rsum[r], off, 32);
      lrun[r] = lrun[r] * corr[r] + rsum[r];
    }
#pragma unroll
    for (int t = 0; t < 4; ++t)
#pragma unroll
      for (int r = 0; r < 8; ++r) o[t][r] *= corr[r];

    // O += P . V : reduction over 64 keys (LDS in-order guarantees P visible)
#pragma unroll
    for (int ks = 0; ks < 64; ks += 4) {
      v2f a = *(const v2f*)&Ps[wave][l16][ks + 2 * half];
#pragma unroll
      for (int t = 0; t < 4; ++t) {
        v2f bf;
        bf.x = Vs[ks + 2 * half][t * 16 + l16];
        bf.y = Vs[ks + 2 * half + 1][t * 16 + l16];
        o[t] = wmma4(a, bf, o[t]);
      }
    }
    __syncthreads();
  }

#pragma unroll
  for (int t = 0; t < 4; ++t)
#pragma unroll
    for (int r = 0; r < 8; ++r) {
      int qrow = q0 + wave * 16 + r + 8 * half;
      Cg[(size_t)qrow * D_MODEL + t * 16 + l16] = o[t][r] / lrun[r];
    }
}

// ---------------------------------------------------------------------------
extern "C" void kernel_launch(void* const* d_in, const int* in_sizes, int n_in,
                              void* d_out, int out_size, void* d_ws, size_t ws_size,
                              hipStream_t stream) {
  const float* query = (const float*)d_in[0];
  const float* key_  = (const float*)d_in[1];
  const float* value = (const float*)d_in[2];
  const int*   mask  = (const int*)d_in[3];
  const float* w_q = (const float*)d_in[4];
  const float* b_q = (const float*)d_in[5];
  const float* w_k = (const float*)d_in[6];
  const float* b_k = (const float*)d_in[7];
  const float* w_v = (const float*)d_in[8];
  const float* b_v = (const float*)d_in[9];
  const float* w_o = (const float*)d_in[10];
  const float* b_o = (const float*)d_in[11];

  const int M = in_sizes[0] / D_MODEL;                       // B*S = 8192
  const int S = (int)((long long)in_sizes[3] / M);           // 2048
  const int B = M / S;                                       // 4

  // Workspace: Q,K,V projections + attention context, each M x D_MODEL f32.
  float* Qp = (float*)d_ws;
  float* Kp = Qp + (size_t)M * D_MODEL;
  float* Vp = Kp + (size_t)M * D_MODEL;
  float* Cp = Vp + (size_t)M * D_MODEL;
  (void)ws_size;

  dim3 gg(M / GM, D_MODEL / GN);
  gemm_xwT_bias_kernel<<<gg, 256, 0, stream>>>(query, w_q, b_q, Qp, M, D_MODEL, D_MODEL);
  gemm_xwT_bias_kernel<<<gg, 256, 0, stream>>>(key_,  w_k, b_k, Kp, M, D_MODEL, D_MODEL);
  gemm_xwT_bias_kernel<<<gg, 256, 0, stream>>>(value, w_v, b_v, Vp, M, D_MODEL, D_MODEL);

  dim3 ga(S / 64, B * N_HEADS);
  flash_attn_kernel<<<ga, 128, 0, stream>>>(Qp, Kp, Vp, mask, Cp, S);

  gemm_xwT_bias_kernel<<<gg, 256, 0, stream>>>(Cp, w_o, b_o, (float*)d_out, M, D_MODEL, D_MODEL);
}